// infoNCELoss_25701084299903
// MI455X (gfx1250) — compile-verified
//
#include <hip/hip_runtime.h>

typedef __attribute__((ext_vector_type(2))) float v2f;
typedef __attribute__((ext_vector_type(8))) float v8f;

#define NROWS 8192
#define BSZ   4096
#define ENC   128
#define INV_T 2.0f      // 1/T, T = 0.5
#define EPSN  1e-8f
#define PSTR  132       // LDS panel row stride (dwords) -> conflict-free b64 reads

// ---------------------------------------------------------------- init
__global__ void nce_init(float* accs) {
    accs[0] = 0.0f;   // sum of lse over all 2B rows
    accs[1] = 0.0f;   // sum of dot(zn_i, zn_{B+i})
}

// ---------------------------------------------------------------- row normalize
__global__ __launch_bounds__(256) void nce_normalize(
        const float* __restrict__ z1, const float* __restrict__ z2,
        float* __restrict__ zn) {
    const int wave = threadIdx.x >> 5;
    const int lane = threadIdx.x & 31;
    const int row  = blockIdx.x * 8 + wave;
    const float* src = (row < BSZ) ? (z1 + (size_t)row * ENC)
                                   : (z2 + (size_t)(row - BSZ) * ENC);
    float4 v = ((const float4*)src)[lane];
    float ss = v.x*v.x + v.y*v.y + v.z*v.z + v.w*v.w;
    #pragma unroll
    for (int m = 16; m; m >>= 1) ss += __shfl_xor(ss, m, 32);
    float inv = 1.0f / fmaxf(sqrtf(ss), EPSN);
    v.x *= inv; v.y *= inv; v.z *= inv; v.w *= inv;
    ((float4*)(zn + (size_t)row * ENC))[lane] = v;
}

// ---------------------------------------------------------------- main: per-row LSE via fp32 WMMA
// grid = NROWS/128 blocks of 256 threads (8 waves); wave w owns 16 rows.
// Each iteration stages a 32-column Zn panel in LDS; each wave runs two
// 16x16 S-tiles against it with dual independent WMMA accumulator chains.
__global__ __launch_bounds__(256) void nce_lse(
        const float* __restrict__ zn, float* __restrict__ lse_acc) {
    __shared__ float panel[32 * PSTR];

    const int lane     = threadIdx.x & 31;
    const int wave     = threadIdx.x >> 5;
    const bool hi      = lane >= 16;
    const int  m       = lane & 15;
    const int  rowBase = blockIdx.x * 128 + wave * 16;
    const int  koff    = hi ? 2 : 0;  // 16x4 f32 A layout: hi half-wave holds K+2,K+3

    // ---- preload this wave's A panel: 16 rows x 128 K as 32 fragments
    v2f a[32];
    const float* arow = zn + (size_t)(rowBase + m) * ENC + koff;
    #pragma unroll
    for (int kf = 0; kf < 32; ++kf) {
        a[kf].x = arow[kf * 4 + 0];
        a[kf].y = arow[kf * 4 + 1];
    }

    float acc[8];
    #pragma unroll
    for (int v = 0; v < 8; ++v) acc[v] = 0.0f;

    for (int jt = 0; jt < NROWS / 32; ++jt) {
        __syncthreads();
        // cooperative load: 32 rows x 128 floats (1024 float4s over 256 threads)
        #pragma unroll
        for (int i = 0; i < 4; ++i) {
            int q  = threadIdx.x + i * 256;     // 0..1023
            int r  = q >> 5;                    // panel row 0..31
            int c4 = q & 31;                    // float4 column
            float4 s = ((const float4*)(zn + (size_t)(jt * 32 + r) * ENC))[c4];
            float* dst = &panel[r * PSTR + c4 * 4];
            dst[0] = s.x; dst[1] = s.y; dst[2] = s.z; dst[3] = s.w;
        }
        __syncthreads();

        #pragma unroll
        for (int half = 0; half < 2; ++half) {
            // preload full B column panel into registers: one wait, then
            // an uninterrupted WMMA burst
            const float* brow = &panel[(half * 16 + m) * PSTR + koff];
            v2f b[32];
            #pragma unroll
            for (int kf = 0; kf < 32; ++kf) {
                b[kf].x = brow[kf * 4 + 0];
                b[kf].y = brow[kf * 4 + 1];
            }
            // dual independent accumulator chains over K
            v8f c0 = {}, c1 = {};
            #pragma unroll
            for (int kf = 0; kf < 32; kf += 2) {
                c0 = __builtin_amdgcn_wmma_f32_16x16x4_f32(
                        false, a[kf + 0], false, b[kf + 0], (short)0, c0, false, false);
                c1 = __builtin_amdgcn_wmma_f32_16x16x4_f32(
                        false, a[kf + 1], false, b[kf + 1], (short)0, c1, false, false);
            }
            // C layout: VGPR v, lanes0-15 -> row v, lanes16-31 -> row v+8
            #pragma unroll
            for (int v = 0; v < 8; ++v) acc[v] += __expf(INV_T * (c0[v] + c1[v]));
        }
    }

    // self-dot per row (exact diagonal exclusion): lane l holds K-subset of row l%16
    float rd = 0.0f;
    #pragma unroll
    for (int kf = 0; kf < 32; ++kf) rd += a[kf].x * a[kf].x + a[kf].y * a[kf].y;
    rd += __shfl_xor(rd, 16, 32);   // lane l: full ||zn_{rowBase + l%16}||^2

    // reduce exp-sums across each 16-lane half
    #pragma unroll
    for (int v = 0; v < 8; ++v) {
        acc[v] += __shfl_xor(acc[v], 8, 32);
        acc[v] += __shfl_xor(acc[v], 4, 32);
        acc[v] += __shfl_xor(acc[v], 2, 32);
        acc[v] += __shfl_xor(acc[v], 1, 32);
    }

    float lsum = 0.0f;
    #pragma unroll
    for (int v = 0; v < 8; ++v) {
        int srcLane = hi ? (v + 8) : v;          // matching row's self-dot
        float rdv = __shfl(rd, srcLane, 32);
        if (m == 0)
            lsum += logf(acc[v] - __expf(INV_T * rdv));
    }
    if (m == 0) atomicAdd(lse_acc, lsum);        // lane 0: rows v, lane 16: rows v+8
}

// ---------------------------------------------------------------- positive pairs
__global__ __launch_bounds__(256) void nce_pos(
        const float* __restrict__ zn, float* __restrict__ pos_acc) {
    const int wave = threadIdx.x >> 5;
    const int lane = threadIdx.x & 31;
    const int i = blockIdx.x * 8 + wave;
    float4 x = ((const float4*)(zn + (size_t)i * ENC))[lane];
    float4 y = ((const float4*)(zn + (size_t)(i + BSZ) * ENC))[lane];
    float d = x.x*y.x + x.y*y.y + x.z*y.z + x.w*y.w;
    #pragma unroll
    for (int m = 16; m; m >>= 1) d += __shfl_xor(d, m, 32);
    if (lane == 0) atomicAdd(pos_acc, d);
}

// ---------------------------------------------------------------- finalize
__global__ void nce_final(const float* __restrict__ accs, float* __restrict__ out) {
    // L = sum lse - sum_i pos_i,  pos_i = 2 * dot_i / T  (counted both directions)
    out[0] = 0.5f * (float)BSZ * (accs[0] - 2.0f * INV_T * accs[1]);
}

extern "C" void kernel_launch(void* const* d_in, const int* in_sizes, int n_in,
                              void* d_out, int out_size, void* d_ws, size_t ws_size,
                              hipStream_t stream) {
    (void)in_sizes; (void)n_in; (void)out_size; (void)ws_size;
    const float* z1 = (const float*)d_in[0];
    const float* z2 = (const float*)d_in[1];
    float* zn   = (float*)d_ws;                  // NROWS*ENC floats (4 MB)
    float* accs = zn + (size_t)NROWS * ENC;      // [lse_sum, pos_sum]
    float* out  = (float*)d_out;

    nce_init     <<<1, 1, 0, stream>>>(accs);
    nce_normalize<<<NROWS / 8, 256, 0, stream>>>(z1, z2, zn);
    nce_lse      <<<NROWS / 128, 256, 0, stream>>>(zn, accs);
    nce_pos      <<<BSZ / 8, 256, 0, stream>>>(zn, accs + 1);
    nce_final    <<<1, 1, 0, stream>>>(accs, out);
}